// RGCN_47562467835963
// MI455X (gfx1250) — compile-verified
//
#include <hip/hip_runtime.h>

#define N_ENT 100000
#define NEDGE 1000000
#define D     100
#define DP    112      // D padded to 7*16 for WMMA column tiles
#define NR    40
#define NB    10
#define EPSV  1e-8f
#define NTC   7        // column tiles of 16 (7*16 = 112)
#define KST   25       // K steps of 4 (25*4 = 100)
#define WELEM (50 * DP * 2)   // 11200 floats per relation, layout [kpair][n][2]
#define NTILES_TOT (N_ENT / 16)   // 6250
#define MAXCHUNK 16384

typedef __attribute__((ext_vector_type(2))) float v2f;
typedef __attribute__((ext_vector_type(8))) float v8f;
typedef __attribute__((ext_vector_type(4))) unsigned int v4u;
typedef __attribute__((ext_vector_type(4))) int v4i;
typedef __attribute__((ext_vector_type(8))) int v8i;

// ---- TDM: stage one relation matrix (11200 contiguous floats) into LDS ----
// D# per CDNA5 ISA ch.8: group0 {count=1, lds_addr, global_addr, type=2},
// group1 {data_size=4B, tensor_dim0=tile_dim0=11200, dim1=1, stride0=11200}.
// This toolchain exposes the 6-arg builtin (g0, g1, g2, g3, g4, cpol).
static __device__ inline void tdm_load_block(const float* gsrc, float* lds_dst) {
  unsigned long long ga = (unsigned long long)(uintptr_t)gsrc;
  unsigned int       la = (unsigned int)(uintptr_t)lds_dst;
  v4u g0;
  g0.x = 1u;                                            // count = 1 valid descriptor
  g0.y = la;                                            // lds_addr
  g0.z = (unsigned int)ga;                              // global_addr[31:0]
  g0.w = (unsigned int)((ga >> 32) & 0x01FFFFFFull)     // global_addr[56:32]
       | 0x80000000u;                                   // type = 2 ("image")
  v8i g1;
  g1[0] = 0x00020000;                      // data_size = 2 (4 bytes)
  g1[1] = (int)(((unsigned)WELEM) << 16);  // tensor_dim0 low16 in [31:16]
  g1[2] = 0x00010000;                      // tensor_dim1 = 1
  g1[3] = (int)(((unsigned)WELEM) << 16);  // tile_dim0
  g1[4] = 1;                               // tile_dim1 = 1
  g1[5] = WELEM;                           // tensor_dim0_stride
  g1[6] = 0;
  g1[7] = 0;
  v4i z4 = {0, 0, 0, 0};
  v8i z8 = {0, 0, 0, 0, 0, 0, 0, 0};
  __builtin_amdgcn_tensor_load_to_lds(g0, g1, z4, z4, z8, 0);
}

// ---------------- utility kernels ----------------

__global__ void k_zero_f(float* p, int n) {
  for (int i = blockIdx.x * blockDim.x + threadIdx.x; i < n; i += gridDim.x * blockDim.x)
    p[i] = 0.f;
}

__global__ void k_zero_i(int* p, int n) {
  for (int i = blockIdx.x * blockDim.x + threadIdx.x; i < n; i += gridDim.x * blockDim.x)
    p[i] = 0;
}

__global__ void k_gather_x0(const int* __restrict__ entity, const float* __restrict__ emb,
                            float* __restrict__ x0, int total) {
  int i = blockIdx.x * blockDim.x + threadIdx.x;
  if (i < total) {
    int n = i / D, d = i - n * D;
    x0[i] = emb[(long)entity[n] * D + d];
  }
}

__global__ void k_edge_prep(const int* __restrict__ src, const int* __restrict__ dst,
                            const int* __restrict__ rel, const float* __restrict__ ew,
                            float* __restrict__ denom, float* __restrict__ cnt,
                            int* __restrict__ rel_cnt) {
  int e = blockIdx.x * blockDim.x + threadIdx.x;
  if (e < NEDGE) {
    int r = rel[e];
    atomicAdd(&denom[(long)src[e] * NR + r], ew[e]);
    atomicAdd(&cnt[dst[e]], 1.0f);
    atomicAdd(&rel_cnt[r], 1);
  }
}

// Serial scan over NR=40 relations: edge offsets + chunk list (8 tiles/chunk).
__global__ void k_scan(const int* __restrict__ rel_cnt, int* __restrict__ rel_off,
                       int* __restrict__ rel_cur, int* __restrict__ chunk_rel,
                       int* __restrict__ chunk_ebase, int* __restrict__ n_chunks) {
  if (blockIdx.x == 0 && threadIdx.x == 0) {
    int ro = 0, nc = 0;
    for (int r = 0; r < NR; r++) {
      rel_off[r] = ro; rel_cur[r] = ro;
      int c = rel_cnt[r];
      int tiles = (c + 15) >> 4;
      for (int t0 = 0; t0 < tiles; t0 += 8) {
        chunk_rel[nc] = r;
        chunk_ebase[nc] = ro + t0 * 16;
        nc++;
      }
      ro += c;
    }
    rel_off[NR] = ro;
    *n_chunks = nc;
  }
}

__global__ void k_normed(const int* __restrict__ src, const int* __restrict__ rel,
                         const float* __restrict__ ew, const float* __restrict__ denom,
                         float* __restrict__ normed) {
  int e = blockIdx.x * blockDim.x + threadIdx.x;
  if (e < NEDGE)
    normed[e] = ew[e] / (denom[(long)src[e] * NR + rel[e]] + EPSV);
}

__global__ void k_bucket(const int* __restrict__ rel, int* __restrict__ rel_cur,
                         int* __restrict__ bucket) {
  int e = blockIdx.x * blockDim.x + threadIdx.x;
  if (e < NEDGE) {
    int pos = atomicAdd(&rel_cur[rel[e]], 1);
    bucket[pos] = e;
  }
}

__global__ void k_invcnt(float* cnt) {
  int n = blockIdx.x * blockDim.x + threadIdx.x;
  if (n < N_ENT) cnt[n] = 1.0f / fmaxf(cnt[n], 1.0f);
}

// w[r] = att[r,:] @ basis, stored per relation as [kpair(50)][n(112)][2] so a
// lane's B pair (k,k+1 at column n) is one aligned 8-byte LDS load.
__global__ void k_compute_w(const float* __restrict__ att, const float* __restrict__ basis,
                            float* __restrict__ wpad) {
  int r = blockIdx.y;
  int idx = blockIdx.x * blockDim.x + threadIdx.x;
  if (idx >= D * DP) return;
  int k = idx / DP, n = idx - k * DP;
  float v = 0.f;
  if (n < D) {
    #pragma unroll
    for (int b = 0; b < NB; b++)
      v += att[r * NB + b] * basis[((long)b * D + k) * D + n];
  }
  wpad[(long)r * WELEM + (((k >> 1) * DP + n) * 2 + (k & 1))] = v;
}

__global__ void k_pad_root(const float* __restrict__ root, float* __restrict__ rootp) {
  int idx = blockIdx.x * blockDim.x + threadIdx.x;
  if (idx < D * DP) {
    int k = idx / DP, n = idx - k * DP;
    rootp[((k >> 1) * DP + n) * 2 + (k & 1)] = (n < D) ? root[k * D + n] : 0.f;
  }
}

// ---------------- main WMMA kernels ----------------
//
// f32 WMMA 16x16x4 operand layout (wave32):
//   A (16x4): lanes 0-15 -> M=lane, K pair {0,1}; lanes 16-31 -> M=lane-16, K pair {2,3}
//   B (4x16): lanes 0-15 -> N=lane, K pair {0,1}; lanes 16-31 -> N=lane-16, K pair {2,3}
//   C/D (16x16): VGPR j, lanes 0-15 -> (M=j, N=lane); lanes 16-31 -> (M=j+8, N=lane-16)

// Persistent blocks pop (relation, 8x16-edge) chunk tickets. Wave 0 stages the
// 44.8KB relation matrix into LDS via the Tensor Data Mover (cached across
// chunks of the same relation); each wave then runs a 16-edge tile:
// msg = X_src(16x100) @ W_r(100x112) as 175 v_wmma_f32_16x16x4_f32, scales rows
// by `normed`, and atomically scatter-adds into agg[dst] (L2-resident).
__global__ void __launch_bounds__(256) k_msg_wmma(
    const float* __restrict__ x, const float* __restrict__ wpad,
    const int* __restrict__ bucket, const int* __restrict__ rel_off,
    const int* __restrict__ chunk_rel, const int* __restrict__ chunk_ebase,
    const int* __restrict__ n_chunks, int* __restrict__ ticket,
    const int* __restrict__ src, const int* __restrict__ dst,
    const float* __restrict__ normed, float* __restrict__ agg)
{
  __shared__ __align__(16) float s_w[WELEM];
  __shared__ int s_info[4];   // [0]=chunk id, [1]=relation, [2]=edge base, [3]=edge end
  __shared__ int s_prev_r;

  const int tid  = threadIdx.x;
  const int lane = tid & 31;
  const int wave = tid >> 5;
  const int m    = lane & 15;
  const int hi   = lane >> 4;
  const int koff = hi * 2;
  const int nchunks = *n_chunks;

  if (tid == 0) s_prev_r = -1;

  for (;;) {
    __syncthreads();   // protect s_w / s_info from previous iteration
    if (tid == 0) {
      int c = atomicAdd(ticket, 1);
      s_info[0] = c;
      if (c < nchunks) {
        int r = chunk_rel[c];
        s_info[1] = r;
        s_info[2] = chunk_ebase[c];
        s_info[3] = rel_off[r + 1];
      }
    }
    __syncthreads();
    if (s_info[0] >= nchunks) break;          // block-uniform exit
    const int r      = s_info[1];
    const int ebase0 = s_info[2];
    const int eend   = s_info[3];

    if (r != s_prev_r) {                      // block-uniform
      if (tid < 32) {                         // wave 0 drives the TDM
        tdm_load_block(wpad + (long)r * WELEM, s_w);
        __builtin_amdgcn_s_wait_tensorcnt(0);
      }
      __syncthreads();
      if (tid == 0) s_prev_r = r;
    }

    int ebase = ebase0 + wave * 16;
    if (ebase < eend) {                       // wave-uniform; EXEC stays full
      int epos  = ebase + m;
      bool valid = epos < eend;
      int e = bucket[valid ? epos : ebase];
      int srow = src[e];
      int drow = dst[e];
      float sc = valid ? normed[e] : 0.f;

      const float* xrow = x + (long)srow * D;

      v8f acc[NTC] = {};
      for (int kt = 0; kt < KST; kt++) {
        int kk = kt * 4 + koff;               // even
        v2f a; a.x = xrow[kk]; a.y = xrow[kk + 1];
        const v2f* bp = (const v2f*)(s_w + (size_t)(kk >> 1) * DP * 2);
        #pragma unroll
        for (int nt = 0; nt < NTC; nt++) {
          v2f b = bp[nt * 16 + m];            // one ds_load_b64, conflict-free
          acc[nt] = __builtin_amdgcn_wmma_f32_16x16x4_f32(
              false, a, false, b, (short)0, acc[nt], false, false);
        }
      }

      #pragma unroll
      for (int nt = 0; nt < NTC; nt++) {
        int col = nt * 16 + m;
        #pragma unroll
        for (int j = 0; j < 8; j++) {
          int rloc = hi * 8 + j;              // owning edge within the 16
          int   dj = __shfl(drow, rloc, 32);
          float sj = __shfl(sc,   rloc, 32);
          if (col < D && sj != 0.f)
            atomicAdd(&agg[(long)dj * D + col], acc[nt][j] * sj);
        }
      }
    }
  }
}

// out = agg*inv_cnt + x@root + bias (+optional ReLU); 8 waves/block, 8 tiles,
// root matrix staged once per block into LDS via TDM.
__global__ void __launch_bounds__(256) k_out_wmma(
    const float* __restrict__ x, const float* __restrict__ rootp,
    const float* __restrict__ agg, const float* __restrict__ invcnt,
    const float* __restrict__ bias, float* __restrict__ out, int relu)
{
  __shared__ __align__(16) float s_w[WELEM];
  const int tid = threadIdx.x;
  if (tid < 32) {
    tdm_load_block(rootp, s_w);
    __builtin_amdgcn_s_wait_tensorcnt(0);
  }
  __syncthreads();

  const int lane = tid & 31;
  const int wave = tid >> 5;
  const int m    = lane & 15;
  const int hi   = lane >> 4;
  const int koff = hi * 2;

  int tile = blockIdx.x * 8 + wave;
  if (tile >= NTILES_TOT) return;             // wave-uniform

  const float* xrow = x + ((long)tile * 16 + m) * D;

  v8f acc[NTC] = {};
  for (int kt = 0; kt < KST; kt++) {
    int kk = kt * 4 + koff;
    v2f a; a.x = xrow[kk]; a.y = xrow[kk + 1];
    const v2f* bp = (const v2f*)(s_w + (size_t)(kk >> 1) * DP * 2);
    #pragma unroll
    for (int nt = 0; nt < NTC; nt++) {
      v2f b = bp[nt * 16 + m];
      acc[nt] = __builtin_amdgcn_wmma_f32_16x16x4_f32(
          false, a, false, b, (short)0, acc[nt], false, false);
    }
  }

  #pragma unroll
  for (int nt = 0; nt < NTC; nt++) {
    int col = nt * 16 + m;
    if (col >= D) continue;
    #pragma unroll
    for (int j = 0; j < 8; j++) {
      long grow = (long)tile * 16 + hi * 8 + j;
      float v = acc[nt][j] + agg[grow * D + col] * invcnt[grow] + bias[col];
      if (relu) v = fmaxf(v, 0.f);
      out[grow * D + col] = v;
    }
  }
}

// ---------------- launch ----------------

extern "C" void kernel_launch(void* const* d_in, const int* in_sizes, int n_in,
                              void* d_out, int out_size, void* d_ws, size_t ws_size,
                              hipStream_t stream) {
  const int*   entity = (const int*)d_in[0];
  const int*   eidx   = (const int*)d_in[1];
  const int*   etype  = (const int*)d_in[2];
  const float* ew     = (const float*)d_in[3];
  const float* emb    = (const float*)d_in[4];
  const float* basis1 = (const float*)d_in[5];
  const float* att1   = (const float*)d_in[6];
  const float* root1  = (const float*)d_in[7];
  const float* bias1  = (const float*)d_in[8];
  const float* basis2 = (const float*)d_in[9];
  const float* att2   = (const float*)d_in[10];
  const float* root2  = (const float*)d_in[11];
  const float* bias2  = (const float*)d_in[12];
  float* out = (float*)d_out;

  const int* src = eidx;
  const int* dst = eidx + NEDGE;

  char* ws = (char*)d_ws;
  size_t off = 0;
  auto alloc = [&](size_t bytes) -> void* {
    void* p = ws + off;
    off += (bytes + 255) & ~(size_t)255;
    return p;
  };
  float* wpad        = (float*)alloc((size_t)NR * WELEM * 4);
  float* rootp       = (float*)alloc((size_t)WELEM * 4);
  float* denom       = (float*)alloc((size_t)N_ENT * NR * 4);
  float* normed      = (float*)alloc((size_t)NEDGE * 4);
  float* cnt         = (float*)alloc((size_t)N_ENT * 4);  // becomes inv_cnt
  int*   rel_cnt     = (int*)alloc(NR * 4);
  int*   rel_off     = (int*)alloc((NR + 1) * 4);
  int*   rel_cur     = (int*)alloc(NR * 4);
  int*   chunk_rel   = (int*)alloc(MAXCHUNK * 4);
  int*   chunk_ebase = (int*)alloc(MAXCHUNK * 4);
  int*   n_chunks    = (int*)alloc(4);
  int*   ticket      = (int*)alloc(4);
  int*   bucket      = (int*)alloc((size_t)NEDGE * 4);
  float* x0          = (float*)alloc((size_t)N_ENT * D * 4);
  float* x1          = (float*)alloc((size_t)N_ENT * D * 4);
  float* agg         = (float*)alloc((size_t)N_ENT * D * 4);

  const int EB = (NEDGE + 255) / 256;
  const int OUTB = (NTILES_TOT + 7) / 8;

  // --- one-time edge-structure prep (shared by both convs) ---
  k_zero_f<<<2048, 256, 0, stream>>>(denom, N_ENT * NR);
  k_zero_f<<<512, 256, 0, stream>>>(cnt, N_ENT);
  k_zero_i<<<1, 64, 0, stream>>>(rel_cnt, NR);
  k_gather_x0<<<(N_ENT * D + 255) / 256, 256, 0, stream>>>(entity, emb, x0, N_ENT * D);
  k_edge_prep<<<EB, 256, 0, stream>>>(src, dst, etype, ew, denom, cnt, rel_cnt);
  k_scan<<<1, 1, 0, stream>>>(rel_cnt, rel_off, rel_cur, chunk_rel, chunk_ebase, n_chunks);
  k_normed<<<EB, 256, 0, stream>>>(src, etype, ew, denom, normed);
  k_bucket<<<EB, 256, 0, stream>>>(etype, rel_cur, bucket);
  k_invcnt<<<(N_ENT + 255) / 256, 256, 0, stream>>>(cnt);

  // --- conv1 (ReLU) ---
  k_compute_w<<<dim3((D * DP + 255) / 256, NR), 256, 0, stream>>>(att1, basis1, wpad);
  k_pad_root<<<(D * DP + 255) / 256, 256, 0, stream>>>(root1, rootp);
  k_zero_f<<<4096, 256, 0, stream>>>(agg, N_ENT * D);
  k_zero_i<<<1, 32, 0, stream>>>(ticket, 1);
  k_msg_wmma<<<1024, 256, 0, stream>>>(x0, wpad, bucket, rel_off, chunk_rel, chunk_ebase,
                                       n_chunks, ticket, src, dst, normed, agg);
  k_out_wmma<<<OUTB, 256, 0, stream>>>(x0, rootp, agg, cnt, bias1, x1, 1);

  // --- conv2 ---
  k_compute_w<<<dim3((D * DP + 255) / 256, NR), 256, 0, stream>>>(att2, basis2, wpad);
  k_pad_root<<<(D * DP + 255) / 256, 256, 0, stream>>>(root2, rootp);
  k_zero_f<<<4096, 256, 0, stream>>>(agg, N_ENT * D);
  k_zero_i<<<1, 32, 0, stream>>>(ticket, 1);
  k_msg_wmma<<<1024, 256, 0, stream>>>(x1, wpad, bucket, rel_off, chunk_rel, chunk_ebase,
                                       n_chunks, ticket, src, dst, normed, agg);
  k_out_wmma<<<OUTB, 256, 0, stream>>>(x1, rootp, agg, cnt, bias2, out, 0);
}